// SingleForgetGateTreeMGU_73684458930390
// MI455X (gfx1250) — compile-verified
//
#include <hip/hip_runtime.h>

typedef float v2f __attribute__((ext_vector_type(2)));
typedef float v8f __attribute__((ext_vector_type(8)));

#define NTHREADS 256   // 8 wave32s
#define TM 16          // nodes per block (GEMM M)
#define HDIM 128
#define KCAT 256       // 2*H concatenated children
#define APAD 260       // LDS row stride for 16x256 tiles (bank rotate, even)
#define XPAD 132       // LDS row stride for 16x128 tiles
#define FPAD 132

__device__ __forceinline__ float sig_(float v) { return 1.0f / (1.0f + __expf(-v)); }

// acc += A(16xK from LDS, per-lane base a0) x B (per-lane base b0).
// Lane L<16 supplies {row, k0..k0+1}; lane L>=16 supplies {row, k0+2..k0+3}
// for both A and B operands, per the CDNA5 16x16x4 f32 VGPR layout.
__device__ __forceinline__ v8f wmma_k(const float* a0, const float* b0, int K, v8f acc) {
#pragma unroll 4
  for (int k = 0; k < K; k += 4) {
    v2f a = *(const v2f*)(a0 + k);
    v2f b = *(const v2f*)(b0 + k);
    acc = __builtin_amdgcn_wmma_f32_16x16x4_f32(false, a, false, b, (short)0, acc,
                                                false, false);
  }
  return acc;
}

// Two GEMMs sharing the same A operand: one ds_load feeds two independent
// accumulator chains (halves DS traffic, interleaves the matrix pipe).
__device__ __forceinline__ void wmma_k2(const float* a0, const float* b0,
                                        const float* b1, int K, v8f& acc0, v8f& acc1) {
#pragma unroll 4
  for (int k = 0; k < K; k += 4) {
    v2f a  = *(const v2f*)(a0 + k);
    v2f p0 = *(const v2f*)(b0 + k);
    v2f p1 = *(const v2f*)(b1 + k);
    acc0 = __builtin_amdgcn_wmma_f32_16x16x4_f32(false, a, false, p0, (short)0, acc0,
                                                 false, false);
    acc1 = __builtin_amdgcn_wmma_f32_16x16x4_f32(false, a, false, p1, (short)0, acc1,
                                                 false, false);
  }
}

// Leaves: h = (1 - sigmoid(wf)) * tanh(whc), with wx computed on the fly.
__global__ __launch_bounds__(NTHREADS)
void mgu_leaf_kernel(float* __restrict__ h, const float* __restrict__ x,
                     const float* __restrict__ Ww, const float* __restrict__ Wb,
                     int start) {
  __shared__ float sX[TM * XPAD];
  const int tid = threadIdx.x;
  const int lane = tid & 31, wave = tid >> 5;
  const int half = lane >> 4, mrow = lane & 15;
  const int n0 = wave * 16;
  const long long node0 = (long long)start + (long long)blockIdx.x * TM;

  const float* xs = x + node0 * HDIM;  // 16 contiguous rows
  for (int e = tid; e < TM * HDIM; e += NTHREADS)
    sX[(e >> 7) * XPAD + (e & 127)] = xs[e];
  __syncthreads();

  const int j = n0 + mrow;
  const float* aX = sX + mrow * XPAD + 2 * half;

  v8f acc_hc = {};  // whc columns j          (W_w rows j)
  v8f acc_f  = {};  // wf  columns j          (W_w rows 128+j)
  wmma_k2(aX, Ww + (size_t)j * HDIM + 2 * half,
              Ww + (size_t)(HDIM + j) * HDIM + 2 * half, HDIM, acc_hc, acc_f);

  const float b_hc = Wb[j], b_f = Wb[HDIM + j];
#pragma unroll
  for (int v = 0; v < 8; ++v) {
    const int m = v + 8 * half;
    const float s = sig_(acc_f[v] + b_f);
    h[(node0 + m) * HDIM + j] = (1.0f - s) * tanhf(acc_hc[v] + b_hc);
  }
}

// Internal level: f = sig(h_cat@Ufᵀ + wf); hc = tanh(whc + (f_rep⊙h_cat)@Uhcᵀ);
// h = f*(c0+c1) + (1-f)*hc, wx fused.
__global__ __launch_bounds__(NTHREADS)
void mgu_internal_kernel(float* __restrict__ h, const float* __restrict__ x,
                         const float* __restrict__ Ww, const float* __restrict__ Wb,
                         const float* __restrict__ Uf, const float* __restrict__ Uhc,
                         int start, int size) {
  __shared__ float sX[TM * XPAD];
  __shared__ float sA[TM * APAD];   // h_cat tile 16x256
  __shared__ float sA2[TM * APAD];  // f_rep ⊙ h_cat
  __shared__ float sF[TM * FPAD];   // gate tile 16x128

  const int tid = threadIdx.x;
  const int lane = tid & 31, wave = tid >> 5;
  const int half = lane >> 4, mrow = lane & 15;
  const int n0 = wave * 16;
  const long long node0 = (long long)start + (long long)blockIdx.x * TM;

  // x tile: 16 contiguous rows; children tile: children of 16 consecutive
  // nodes are 32 consecutive nodes -> one contiguous 16KB region of h.
  const float* xs = x + node0 * HDIM;
  const float* hs = h + (2 * node0 + 1) * HDIM;
  for (int e = tid; e < TM * HDIM; e += NTHREADS)
    sX[(e >> 7) * XPAD + (e & 127)] = xs[e];
  for (int e = tid; e < TM * KCAT; e += NTHREADS)
    sA[(e >> 8) * APAD + (e & 255)] = hs[e];
  __syncthreads();

  const int j = n0 + mrow;
  const float* aX = sX + mrow * XPAD + 2 * half;
  const float* aA = sA + mrow * APAD + 2 * half;

  // Fused W-GEMM pair over the shared x tile:
  //   whc = x@Wwᵀ cols j ; wf = x@Wwᵀ cols 128+j
  v8f acc_whc = {};
  v8f acc_wf  = {};
  wmma_k2(aX, Ww + (size_t)j * HDIM + 2 * half,
              Ww + (size_t)(HDIM + j) * HDIM + 2 * half, HDIM, acc_whc, acc_wf);

  // f-pre = h_cat@Ufᵀ cols j
  v8f acc_f = {};
  acc_f = wmma_k(aA, Uf + (size_t)j * KCAT + 2 * half, KCAT, acc_f);

  const float b_f = Wb[HDIM + j], b_hc = Wb[j];
  v8f fgate;
#pragma unroll
  for (int v = 0; v < 8; ++v) fgate[v] = sig_(acc_f[v] + acc_wf[v] + b_f);

  // share f across waves (each wave only owns 16 of the 128 gate columns)
#pragma unroll
  for (int v = 0; v < 8; ++v) sF[(v + 8 * half) * FPAD + j] = fgate[v];
  __syncthreads();
  for (int e = tid; e < TM * KCAT; e += NTHREADS) {
    const int m = e >> 8, k = e & 255;
    sA2[m * APAD + k] = sF[m * FPAD + (k & 127)] * sA[m * APAD + k];
  }
  __syncthreads();

  // cand-pre = A2@Uhcᵀ cols j
  v8f acc_hc = {};
  acc_hc = wmma_k(sA2 + mrow * APAD + 2 * half, Uhc + (size_t)j * KCAT + 2 * half,
                  KCAT, acc_hc);

#pragma unroll
  for (int v = 0; v < 8; ++v) {
    const int m = v + 8 * half;
    const float cs = sA[m * APAD + j] + sA[m * APAD + HDIM + j];  // child sum
    const float hc = tanhf(acc_whc[v] + b_hc + acc_hc[v]);
    const float fv = fgate[v];
    const long long node = node0 + m;
    if (node < (long long)start + size)  // guard partial top-level tiles
      h[node * HDIM + j] = fv * cs + (1.0f - fv) * hc;
  }
}

extern "C" void kernel_launch(void* const* d_in, const int* in_sizes, int n_in,
                              void* d_out, int out_size, void* d_ws, size_t ws_size,
                              hipStream_t stream) {
  const float* x   = (const float*)d_in[0];
  const float* Ww  = (const float*)d_in[1];
  const float* Wb  = (const float*)d_in[2];
  const float* Uf  = (const float*)d_in[3];
  const float* Uhc = (const float*)d_in[4];
  float* h = (float*)d_out;

  const int n = in_sizes[0] / HDIM;        // 2^depth - 1
  int depth = 0;
  while ((1 << (depth + 1)) <= n + 1) ++depth;  // n+1 = 2^depth -> depth

  // Leaves (level depth-1): count = 2^(depth-1), divisible by 16 for depth>=5.
  const int leafStart = (1 << (depth - 1)) - 1;
  const int leafCount = 1 << (depth - 1);
  mgu_leaf_kernel<<<leafCount / TM, NTHREADS, 0, stream>>>(h, x, Ww, Wb, leafStart);

  // Internal levels, bottom-up, stream-ordered (strict dependency chain).
  for (int lvl = depth - 2; lvl >= 0; --lvl) {
    const int size = 1 << lvl;
    const int start = size - 1;
    const int blocks = (size + TM - 1) / TM;
    mgu_internal_kernel<<<blocks, NTHREADS, 0, stream>>>(h, x, Ww, Wb, Uf, Uhc,
                                                         start, size);
  }
}